// NeuralODEProcess_14714557956419
// MI455X (gfx1250) — compile-verified
//
#include <hip/hip_runtime.h>
#include <math.h>

// ---------------------------------------------------------------------------
// Neural ODE Process forward for MI455X (gfx1250), fp32 throughout.
// All GEMMs run on V_WMMA_F32_16X16X4_F32 (wave32 matrix pipe).
// Compute-bound (~280 GFLOP fp32 vs ~6MB of weights resident in 192MB L2),
// so per-stage fusion + LDS staging + WMMA is the optimal structure; HBM
// traffic is only the 34MB of intermediates + 8MB output (<<1% of roofline).
// ---------------------------------------------------------------------------

#define B_SZ  256
#define NCtx  64
#define NT    256
#define YD    16
#define RD    256
#define ZD    128
#define HD    512
#define LD    64

typedef float v2f __attribute__((ext_vector_type(2)));
typedef float v8f __attribute__((ext_vector_type(8)));

// One 16x16 output tile accumulation: acc += A(16xK from LDS) * W^T tile.
// W is row-major (Nout x ldw); B-matrix element (k,n) = W[n0+n][wk0+k].
// ISA 7.12.2 f32 layouts: A lane(0..15)=M, VGPR0={K0|K2}, VGPR1={K1|K3}
// (lanes 16-31 take the K+2 pair); B symmetric with N across lanes.
__device__ __forceinline__ void wmma_accum(v8f& acc,
    const float* in, int lda,
    const float* __restrict__ W, int ldw, int wk0,
    int n0, int Kloop, int Kreal)
{
  const int lane = threadIdx.x & 31;
  const int half = lane >> 4;
  const int l15  = lane & 15;
  const float* __restrict__ wr = W + (size_t)(n0 + l15) * (size_t)ldw + wk0;
  const float* ar = in + l15 * lda;
  __builtin_prefetch(wr, 0, 1);               // global_prefetch_b8
  for (int k0 = 0; k0 < Kloop; k0 += 4) {
    const int ka = k0 + (half << 1);
    v2f a, b;
    a.x = ar[ka];
    a.y = ar[ka + 1];
    b.x = (ka     < Kreal) ? wr[ka]     : 0.f;   // guard ragged K tail
    b.y = (ka + 1 < Kreal) ? wr[ka + 1] : 0.f;
    acc = __builtin_amdgcn_wmma_f32_16x16x4_f32(
        false, a, false, b, (short)0, acc, false, false);
  }
}

// Block-level layer: out(16 x Nout) = act(in(16 x K) @ W^T + bias).
// 8 waves, each wave owns 16x16 tiles strided by 8. act: 0=none 1=relu 2=tanh
__device__ __forceinline__ void mlp_layer(const float* in, int lda,
    const float* __restrict__ W, int ldw,
    const float* __restrict__ bias,
    float* out, int ldd, int Nout, int Kloop, int Kreal, int act)
{
  const int wave = threadIdx.x >> 5;
  const int lane = threadIdx.x & 31;
  const int half = lane >> 4, l15 = lane & 15;
  for (int nt = wave; nt < (Nout >> 4); nt += 8) {
    const int n0 = nt << 4;
    v8f acc;
#pragma unroll
    for (int rr = 0; rr < 8; ++rr) acc[rr] = 0.f;
    wmma_accum(acc, in, lda, W, ldw, 0, n0, Kloop, Kreal);
    const float bv = bias[n0 + l15];
#pragma unroll
    for (int rr = 0; rr < 8; ++rr) {
      float v = acc[rr] + bv;
      if (act == 1)      v = fmaxf(v, 0.f);
      else if (act == 2) v = tanhf(v);
      out[(size_t)(rr + 8 * half) * ldd + n0 + l15] = v;
    }
  }
  __syncthreads();
}

// --------------------------- encoder ---------------------------------------
// rows = B*NC; 17 -> 512 -> 512 -> 256 (relu,relu,none) ; writes R_i global.
__global__ void __launch_bounds__(256) enc_kernel(
    const float* __restrict__ xc, const float* __restrict__ yc,
    const float* __restrict__ w1, const float* __restrict__ b1,
    const float* __restrict__ w2, const float* __restrict__ b2,
    const float* __restrict__ w3, const float* __restrict__ b3,
    float* __restrict__ Ri)
{
  __shared__ float in0[16 * 20];     // 17 padded to 20
  __shared__ float hA[16 * HD];
  __shared__ float hB[16 * HD];
  const int t = threadIdx.x;
  const int r0 = blockIdx.x * 16;
  for (int i = t; i < 16 * 20; i += 256) {
    int r = i / 20, c = i % 20;
    size_t row = (size_t)(r0 + r);
    float v;
    if (c == 0)      v = xc[row];
    else if (c < 17) v = yc[row * 16 + (c - 1)];
    else             v = 0.f;
    in0[i] = v;
  }
  __syncthreads();
  mlp_layer(in0, 20,  w1, 17,  b1, hA, HD, HD, 20,  17,  1);
  mlp_layer(hA,  HD,  w2, HD,  b2, hB, HD, HD, HD,  HD,  1);
  mlp_layer(hB,  HD,  w3, HD,  b3, Ri + (size_t)r0 * RD, RD, RD, HD, HD, 0);
}

// mean over NC context points: r[b][j] = mean_i Ri[b*NC+i][j]
__global__ void __launch_bounds__(256) mean_kernel(
    const float* __restrict__ Ri, float* __restrict__ r)
{
  const int b = blockIdx.x, j = threadIdx.x;
  float s = 0.f;
  for (int i = 0; i < NCtx; ++i) s += Ri[((size_t)b * NCtx + i) * RD + j];
  r[(size_t)b * RD + j] = s * (1.f / NCtx);
}

// --------------------------- latent head -----------------------------------
// hid = relu(r@wh^T+bh); mu = hid@wmu^T; sig = 0.1+0.9*sigmoid(hid@wsig^T);
// z = mu + sig*eps  (z: B x 128)
__global__ void __launch_bounds__(256) ms_kernel(
    const float* __restrict__ rg,
    const float* __restrict__ wh,  const float* __restrict__ bh,
    const float* __restrict__ wmu, const float* __restrict__ bmu,
    const float* __restrict__ wsg, const float* __restrict__ bsg,
    const float* __restrict__ eps, float* __restrict__ z)
{
  __shared__ float rin[16 * RD];
  __shared__ float hid[16 * RD];
  __shared__ float msb[16 * RD];   // cols 0..127 = mu, 128..255 = pre-sigma
  const int t = threadIdx.x;
  const int r0 = blockIdx.x * 16;
  for (int i = t; i < 16 * RD; i += 256) rin[i] = rg[(size_t)r0 * RD + i];
  __syncthreads();
  mlp_layer(rin, RD, wh,  RD, bh,  hid,       RD, RD, RD, RD, 1);
  mlp_layer(hid, RD, wmu, RD, bmu, msb,       RD, ZD, RD, RD, 0);
  mlp_layer(hid, RD, wsg, RD, bsg, msb + ZD,  RD, ZD, RD, RD, 0);
  for (int i = t; i < 16 * ZD; i += 256) {
    int r = i >> 7, c = i & 127;
    float mu = msb[r * RD + c];
    float pg = msb[r * RD + ZD + c];
    float sg = 0.1f + 0.9f / (1.f + expf(-pg));
    size_t row = (size_t)(r0 + r);
    z[row * ZD + c] = mu + sg * eps[row * ZD + c];
  }
}

// --------------------------- ODE integrator --------------------------------
// 16 blocks x 16 batch rows; RK4, one step per target interval (dt ~ 1/256,
// smooth tanh field -> well inside dopri5's step size at rtol=1e-7).
// vin row layout: [L(64) | z_frozen(64) | t | pad(3)] (K=129 padded to 132)
__device__ __forceinline__ void ode_feval(
    const float* vin, float* h1, float* h2, float* dL,
    const float* w1, const float* b1, const float* w2, const float* b2,
    const float* w3, const float* b3)
{
  mlp_layer(vin, 132, w1, 129, b1, h1, HD, HD, 132, 129, 2);
  mlp_layer(h1,  HD,  w2, HD,  b2, h2, HD, HD, HD,  HD,  2);
  mlp_layer(h2,  HD,  w3, HD,  b3, dL, LD, LD, HD,  HD,  0);
}

__global__ void __launch_bounds__(256) ode_kernel(
    const float* __restrict__ z, const float* __restrict__ xt,
    const float* __restrict__ w1, const float* __restrict__ b1,
    const float* __restrict__ w2, const float* __restrict__ b2,
    const float* __restrict__ w3, const float* __restrict__ b3,
    float* __restrict__ latent)
{
  __shared__ float vin[16 * 132];
  __shared__ float h1 [16 * HD];
  __shared__ float h2 [16 * HD];
  __shared__ float dL [16 * LD];
  __shared__ float Lst[16 * LD];
  __shared__ float acc[16 * LD];
  const int t = threadIdx.x;
  const int r0 = blockIdx.x * 16;

  for (int i = t; i < 16 * LD; i += 256) {
    int r = i >> 6, c = i & 63;
    size_t row = (size_t)(r0 + r);
    Lst[i] = z[row * ZD + c];                 // evolving part
    vin[r * 132 + LD + c] = z[row * ZD + LD + c]; // frozen part (constant)
  }
  for (int i = t; i < 16 * 3; i += 256) {     // zero K-padding columns
    int r = i / 3, c = i % 3;
    vin[r * 132 + 129 + c] = 0.f;
  }
  __syncthreads();

  float tprev = 0.f;
  for (int step = 0; step < NT; ++step) {
    const float t1 = xt[step];
    const float h  = t1 - tprev;
    const float tm = tprev + 0.5f * h;

    // k1 = f(L, t0)
    for (int i = t; i < 16 * LD; i += 256) {
      int r = i >> 6, c = i & 63;
      vin[r * 132 + c] = Lst[i];
    }
    if (t < 16) vin[t * 132 + 128] = tprev;
    __syncthreads();
    ode_feval(vin, h1, h2, dL, w1, b1, w2, b2, w3, b3);

    // k2 = f(L + h/2*k1, tm)
    for (int i = t; i < 16 * LD; i += 256) {
      int r = i >> 6, c = i & 63;
      float k = dL[i];
      acc[i] = k;
      vin[r * 132 + c] = Lst[i] + 0.5f * h * k;
    }
    if (t < 16) vin[t * 132 + 128] = tm;
    __syncthreads();
    ode_feval(vin, h1, h2, dL, w1, b1, w2, b2, w3, b3);

    // k3 = f(L + h/2*k2, tm)
    for (int i = t; i < 16 * LD; i += 256) {
      int r = i >> 6, c = i & 63;
      float k = dL[i];
      acc[i] += 2.f * k;
      vin[r * 132 + c] = Lst[i] + 0.5f * h * k;
    }
    if (t < 16) vin[t * 132 + 128] = tm;
    __syncthreads();
    ode_feval(vin, h1, h2, dL, w1, b1, w2, b2, w3, b3);

    // k4 = f(L + h*k3, t1)
    for (int i = t; i < 16 * LD; i += 256) {
      int r = i >> 6, c = i & 63;
      float k = dL[i];
      acc[i] += 2.f * k;
      vin[r * 132 + c] = Lst[i] + h * k;
    }
    if (t < 16) vin[t * 132 + 128] = t1;
    __syncthreads();
    ode_feval(vin, h1, h2, dL, w1, b1, w2, b2, w3, b3);

    for (int i = t; i < 16 * LD; i += 256) {
      int r = i >> 6, c = i & 63;
      float L = Lst[i] + (h * (1.f / 6.f)) * (acc[i] + dL[i]);
      Lst[i] = L;
      latent[((size_t)(r0 + r) * NT + step) * LD + c] = L;
    }
    __syncthreads();
    tprev = t1;
  }
}

// --------------------------- decoder + output heads ------------------------
// rows = B*NT; trip = [x(1)|latent(64)|z_hi(64)] -> 512 -> 512 -> 512 (relu)
// hid2 = [latent(64)|h(512)]; mu = hid2@hm^T+b ; sigma = 0.1+0.9*softplus(.)
__global__ void __launch_bounds__(256) dec_kernel(
    const float* __restrict__ xt, const float* __restrict__ latent,
    const float* __restrict__ z,
    const float* __restrict__ w1, const float* __restrict__ b1,
    const float* __restrict__ w2, const float* __restrict__ b2,
    const float* __restrict__ w3, const float* __restrict__ b3,
    const float* __restrict__ hmw, const float* __restrict__ hmb,
    const float* __restrict__ hsw, const float* __restrict__ hsb,
    float* __restrict__ out)
{
  __shared__ float in0[16 * 132];
  __shared__ float hA [16 * HD];
  __shared__ float hB [16 * HD];
  const int t = threadIdx.x;
  const int r0 = blockIdx.x * 16;
  for (int i = t; i < 16 * 132; i += 256) {
    int r = i / 132, c = i % 132;
    size_t row = (size_t)(r0 + r);
    float v;
    if (c == 0)       v = xt[row];                       // X_DIM == 1
    else if (c < 65)  v = latent[row * LD + (c - 1)];
    else if (c < 129) { size_t b = row >> 8; v = z[b * ZD + LD + (c - 65)]; }
    else              v = 0.f;
    in0[i] = v;
  }
  __syncthreads();
  mlp_layer(in0, 132, w1, 129, b1, hA, HD, HD, 132, 129, 1);
  mlp_layer(hA,  HD,  w2, HD,  b2, hB, HD, HD, HD,  HD,  1);
  mlp_layer(hB,  HD,  w3, HD,  b3, hA, HD, HD, HD,  HD,  1);

  const int wave = t >> 5, lane = t & 31, half = lane >> 4, l15 = lane & 15;
  if (wave < 2) {                       // wave0: mu head, wave1: sigma head
    const float* W  = (wave == 0) ? hmw : hsw;
    const float* bb = (wave == 0) ? hmb : hsb;
    v8f accv;
#pragma unroll
    for (int rr = 0; rr < 8; ++rr) accv[rr] = 0.f;
    // hid2 split: latent part lives in in0 cols 1..64, h part in hA
    wmma_accum(accv, in0 + 1, 132, W, 576, 0,  0, 64,  64);
    wmma_accum(accv, hA,      HD,  W, 576, 64, 0, HD,  HD);
    const float bv = bb[l15];
    const size_t base = (wave == 0) ? 0 : (size_t)B_SZ * NT * YD;
#pragma unroll
    for (int rr = 0; rr < 8; ++rr) {
      float v = accv[rr] + bv;
      if (wave == 1) {
        float sp = (v > 20.f) ? v : log1pf(expf(v));
        v = 0.1f + 0.9f * sp;
      }
      size_t row = (size_t)(r0 + rr + 8 * half);
      out[base + row * YD + l15] = v;
    }
  }
}

// --------------------------- launch ----------------------------------------
extern "C" void kernel_launch(void* const* d_in, const int* in_sizes, int n_in,
                              void* d_out, int out_size, void* d_ws, size_t ws_size,
                              hipStream_t stream) {
  const float* x_context = (const float*)d_in[0];
  const float* y_context = (const float*)d_in[1];
  const float* x_target  = (const float*)d_in[2];
  const float* eps       = (const float*)d_in[3];
  const float* enc_w1 = (const float*)d_in[4];  const float* enc_b1 = (const float*)d_in[5];
  const float* enc_w2 = (const float*)d_in[6];  const float* enc_b2 = (const float*)d_in[7];
  const float* enc_w3 = (const float*)d_in[8];  const float* enc_b3 = (const float*)d_in[9];
  const float* ms_wh  = (const float*)d_in[10]; const float* ms_bh  = (const float*)d_in[11];
  const float* ms_wmu = (const float*)d_in[12]; const float* ms_bmu = (const float*)d_in[13];
  const float* ms_wsg = (const float*)d_in[14]; const float* ms_bsg = (const float*)d_in[15];
  const float* ode_w1 = (const float*)d_in[16]; const float* ode_b1 = (const float*)d_in[17];
  const float* ode_w2 = (const float*)d_in[18]; const float* ode_b2 = (const float*)d_in[19];
  const float* ode_w3 = (const float*)d_in[20]; const float* ode_b3 = (const float*)d_in[21];
  const float* dec_w1 = (const float*)d_in[22]; const float* dec_b1 = (const float*)d_in[23];
  const float* dec_w2 = (const float*)d_in[24]; const float* dec_b2 = (const float*)d_in[25];
  const float* dec_w3 = (const float*)d_in[26]; const float* dec_b3 = (const float*)d_in[27];
  const float* hm_w   = (const float*)d_in[28]; const float* hm_b   = (const float*)d_in[29];
  const float* hs_w   = (const float*)d_in[30]; const float* hs_b   = (const float*)d_in[31];

  float* ws     = (float*)d_ws;
  float* Ri     = ws;                                    // (B*NC) x 256
  float* rbuf   = Ri   + (size_t)B_SZ * NCtx * RD;       // B x 256
  float* zbuf   = rbuf + (size_t)B_SZ * RD;              // B x 128
  float* latent = zbuf + (size_t)B_SZ * ZD;              // B x NT x 64

  enc_kernel<<<(B_SZ * NCtx) / 16, 256, 0, stream>>>(
      x_context, y_context, enc_w1, enc_b1, enc_w2, enc_b2, enc_w3, enc_b3, Ri);
  mean_kernel<<<B_SZ, RD, 0, stream>>>(Ri, rbuf);
  ms_kernel<<<B_SZ / 16, 256, 0, stream>>>(
      rbuf, ms_wh, ms_bh, ms_wmu, ms_bmu, ms_wsg, ms_bsg, eps, zbuf);
  ode_kernel<<<B_SZ / 16, 256, 0, stream>>>(
      zbuf, x_target, ode_w1, ode_b1, ode_w2, ode_b2, ode_w3, ode_b3, latent);
  dec_kernel<<<(B_SZ * NT) / 16, 256, 0, stream>>>(
      x_target, latent, zbuf, dec_w1, dec_b1, dec_w2, dec_b2, dec_w3, dec_b3,
      hm_w, hm_b, hs_w, hs_b, (float*)d_out);
}